// crf_12919261626812
// MI455X (gfx1250) — compile-verified
//
#include <hip/hip_runtime.h>
#include <stdint.h>

// Problem constants (match reference: B=8192, L=2048, 2 states)
#define CRF_B      8192
#define CRF_L      2048
#define CRF_NCHUNK 16          // chunks per row (log-semiring parallel scan)
#define CRF_CLEN   128         // steps per chunk
#define CRF_WSTEPS 4           // steps staged per LDS window
#define CRF_NWIN   32          // CLEN / WSTEPS
#define CRF_RPB    16          // rows per block
#define CRF_TPB    256         // threads per block = RPB * NCHUNK

// LDS layout (per double buffer). All async B128 destinations 16B aligned.
#define CRF_EM_STRIDE 48                       // 32B data + pad (bank spread)
#define CRF_EM_OFF    0
#define CRF_EM_SZ     (CRF_TPB * CRF_EM_STRIDE)      // 12288
#define CRF_LAB_OFF   CRF_EM_SZ                      // 12288
#define CRF_WHO_OFF   (CRF_LAB_OFF + CRF_TPB * 16)   // 16384
#define CRF_POS_OFF   (CRF_WHO_OFF + CRF_TPB * 16)   // 20480
#define CRF_BUF_SZ    (CRF_POS_OFF + CRF_TPB * 16)   // 24576
#define CRF_COMBO_OFF (2 * CRF_BUF_SZ)               // 49152
#define CRF_SMEM_SZ   (CRF_COMBO_OFF + 60 * 4 * 4)   // 50112 bytes

__device__ __forceinline__ float lse2f(float a, float b) {
  float mx = fmaxf(a, b);
  float d  = fabsf(a - b);
  return mx + log1pf(__expf(-d));
}

// CDNA5 async global->LDS DMA (VGLOBAL encoding, GV mode, tracked by ASYNCcnt)
__device__ __forceinline__ void async_b128(uint32_t lds_addr, const void* gaddr) {
  asm volatile("global_load_async_to_lds_b128 %0, %1, off"
               :: "v"(lds_addr), "v"(gaddr)
               : "memory");
}

__global__ __launch_bounds__(CRF_TPB)
void crf_chunk_kernel(const float* __restrict__ em,
                      const float* __restrict__ whoP,   // (2,2,2)
                      const float* __restrict__ posP,   // (19,2,2)
                      const int*   __restrict__ lab,
                      const int*   __restrict__ who,
                      const int*   __restrict__ pos,
                      float* __restrict__ wsM,          // [B*NCHUNK*4]
                      float* __restrict__ wsG)          // [B*NCHUNK]
{
  __shared__ __align__(16) unsigned char smem[CRF_SMEM_SZ];

  const int t   = threadIdx.x;
  const int r   = t >> 4;        // row within block
  const int c   = t & 15;        // chunk index
  const int row = blockIdx.x * CRF_RPB + r;
  const size_t rowL = (size_t)row * CRF_L;
  const int l0  = c * CRF_CLEN;

  const float* emRow  = em  + (rowL + l0) * 2;
  const int*   labRow = lab + rowL + l0;
  const int*   whoRow = who + rowL + l0;
  const int*   posRow = pos + rowL + l0;

  // Combined transition bank: combo[who*20 + pos][i*2+j] = who_bank + pos_bank
  // (who==2 and pos==19 are the zero pads of the reference).
  float* combo = (float*)(smem + CRF_COMBO_OFF);
  for (int i = t; i < 60 * 4; i += CRF_TPB) {
    int idx = i >> 2, comp = i & 3;
    int wi = idx / 20, pi = idx % 20;
    float v = 0.f;
    if (wi < 2)  v += whoP[wi * 4 + comp];
    if (pi < 19) v += posP[pi * 4 + comp];
    combo[i] = v;
  }

  // Low 32 bits of the generic address of a __shared__ object == LDS byte offset.
  const uint32_t ldsBase = (uint32_t)(uintptr_t)&smem[0];
  const uint32_t emL0  = ldsBase + CRF_EM_OFF  + t * CRF_EM_STRIDE;
  const uint32_t labL0 = ldsBase + CRF_LAB_OFF + t * 16;
  const uint32_t whoL0 = ldsBase + CRF_WHO_OFF + t * 16;
  const uint32_t posL0 = ldsBase + CRF_POS_OFF + t * 16;

  // Issue one window's worth of async DMA (5 x B128 per thread -> ASYNCcnt += 5/wave)
  #define CRF_ISSUE(w_, b_) do {                                   \
    int sb = (w_) * CRF_WSTEPS;                                    \
    uint32_t bo = (uint32_t)(b_) * CRF_BUF_SZ;                     \
    async_b128(emL0  + bo,      emRow + sb * 2);                   \
    async_b128(emL0  + bo + 16, emRow + sb * 2 + 4);               \
    async_b128(labL0 + bo,      labRow + sb);                      \
    async_b128(whoL0 + bo,      whoRow + sb);                      \
    async_b128(posL0 + bo,      posRow + sb);                      \
  } while (0)

  CRF_ISSUE(0, 0);

  // Log-semiring identity matrix; -1e30 acts as -inf without NaN hazards.
  float m00 = 0.f, m01 = -1e30f, m10 = -1e30f, m11 = 0.f;
  float gsum = 0.f;
  int plab = (c > 0) ? labRow[-1] : 0;   // one stray scalar load per thread

  for (int w = 0; w < CRF_NWIN; ++w) {
    const int buf = w & 1;
    if (w + 1 < CRF_NWIN) {
      CRF_ISSUE(w + 1, (w + 1) & 1);
      asm volatile("s_wait_asynccnt 0x5" ::: "memory");  // window w complete
    } else {
      asm volatile("s_wait_asynccnt 0x0" ::: "memory");
    }
    __syncthreads();

    const uint32_t bo = (uint32_t)buf * CRF_BUF_SZ;
    const float* emS  = (const float*)(smem + bo + CRF_EM_OFF  + t * CRF_EM_STRIDE);
    const int*   labS = (const int*)  (smem + bo + CRF_LAB_OFF + t * 16);
    const int*   whoS = (const int*)  (smem + bo + CRF_WHO_OFF + t * 16);
    const int*   posS = (const int*)  (smem + bo + CRF_POS_OFF + t * 16);

    #pragma unroll
    for (int s = 0; s < CRF_WSTEPS; ++s) {
      float e0 = emS[2 * s], e1 = emS[2 * s + 1];
      int lb = labS[s], wi = whoS[s], pi = posS[s];
      const float4 tv = *(const float4*)&combo[(wi * 20 + pi) * 4];
      float t00 = tv.x, t01 = tv.y, t10 = tv.z, t11 = tv.w;

      // gold score partials
      gsum += lb ? e1 : e0;
      int l = l0 + w * CRF_WSTEPS + s;
      if (l > 0)
        gsum += plab ? (lb ? t11 : t10) : (lb ? t01 : t00);
      plab = lb;

      // M <- M (x) S_t, S_t[i][j] = T[i][j] + e[j]   (logsumexp semiring)
      float n00 = lse2f(m00 + t00, m01 + t10) + e0;
      float n01 = lse2f(m00 + t01, m01 + t11) + e1;
      float n10 = lse2f(m10 + t00, m11 + t10) + e0;
      float n11 = lse2f(m10 + t01, m11 + t11) + e1;
      m00 = n00; m01 = n01; m10 = n10; m11 = n11;
    }
    __syncthreads();   // protect buffer w&1 before it is re-issued
  }

  const int chunkId = row * CRF_NCHUNK + c;
  float4 mv; mv.x = m00; mv.y = m01; mv.z = m10; mv.w = m11;
  *(float4*)&wsM[chunkId * 4] = mv;
  wsG[chunkId] = gsum;
  #undef CRF_ISSUE
}

__global__ __launch_bounds__(256)
void crf_combine_kernel(const float* __restrict__ wsM,
                        const float* __restrict__ wsG,
                        float* __restrict__ out)
{
  int row = blockIdx.x * blockDim.x + threadIdx.x;
  if (row >= CRF_B) return;
  float a0 = 0.f, a1 = 0.f, g = 0.f;
  #pragma unroll 4
  for (int c = 0; c < CRF_NCHUNK; ++c) {
    const float4 m = *(const float4*)&wsM[(row * CRF_NCHUNK + c) * 4];
    float n0 = lse2f(a0 + m.x, a1 + m.z);
    float n1 = lse2f(a0 + m.y, a1 + m.w);
    a0 = n0; a1 = n1;
    g += wsG[row * CRF_NCHUNK + c];
  }
  out[row]         = g;              // gold_score
  out[CRF_B + row] = lse2f(a0, a1);  // total_score
}

extern "C" void kernel_launch(void* const* d_in, const int* in_sizes, int n_in,
                              void* d_out, int out_size, void* d_ws, size_t ws_size,
                              hipStream_t stream) {
  (void)in_sizes; (void)n_in; (void)out_size; (void)ws_size;
  const float* em   = (const float*)d_in[0];  // (B, L, 2)
  const float* whoP = (const float*)d_in[1];  // (2, 2, 2)
  const float* posP = (const float*)d_in[2];  // (19, 2, 2)
  const int*   lab  = (const int*)d_in[3];    // (B, L)
  const int*   who  = (const int*)d_in[4];    // (B, L)
  const int*   pos  = (const int*)d_in[5];    // (B, L)
  // d_in[6] prior, d_in[7] posterior_sequence: unused by the reference math.

  float* wsM = (float*)d_ws;                              // B*NCHUNK*4 floats
  float* wsG = wsM + (size_t)CRF_B * CRF_NCHUNK * 4;      // B*NCHUNK floats
  float* out = (float*)d_out;                             // 2*B floats

  crf_chunk_kernel<<<CRF_B / CRF_RPB, CRF_TPB, 0, stream>>>(
      em, whoP, posP, lab, who, pos, wsM, wsG);
  crf_combine_kernel<<<CRF_B / 256, 256, 0, stream>>>(wsM, wsG, out);
}